// GraphEncoder_43344809951367
// MI455X (gfx1250) — compile-verified
//
#include <hip/hip_runtime.h>
#include <hip/hip_bf16.h>
#include <stdint.h>

typedef __attribute__((ext_vector_type(2))) float v2f;
typedef __attribute__((ext_vector_type(8))) float v8f;

#define NEG_SLOPE 0.2f

// ---- ordered-uint encoding of float for atomicMax-based segment_max ----
__device__ __forceinline__ unsigned enc_f32(float f) {
    unsigned u = __float_as_uint(f);
    return (u & 0x80000000u) ? ~u : (u | 0x80000000u);
}
__device__ __forceinline__ float dec_f32(unsigned u) {
    unsigned b = (u & 0x80000000u) ? (u ^ 0x80000000u) : ~u;
    return __uint_as_float(b);
}

// ---------------------------------------------------------------------------
// init: out[n][f] = bias[f]; denom = 0; emax = enc-identity (0)
// ---------------------------------------------------------------------------
__global__ void gat_init_kernel(float* __restrict__ out, const float* __restrict__ bias,
                                float* __restrict__ denom, unsigned* __restrict__ emax,
                                int N) {
    long long idx = (long long)blockIdx.x * blockDim.x + threadIdx.x;
    if (idx < (long long)N * 64) out[idx] = bias[idx & 63];
    if (idx < N) { denom[idx] = 0.0f; emax[idx] = 0u; }  // 0 < enc(-inf)=0x007FFFFF
}

// ---------------------------------------------------------------------------
// h = x @ W via V_WMMA_F32_16X16X4_F32.
// Block = 256 threads = 8 waves. Wave w computes rows [16*tile, 16*tile+16) x 64 cols.
// W (64x64 f32, 16KB) staged in LDS once per block.
// A frag (16x4 f32): lane<16 -> (M=lane, K=k0,k0+1); lane>=16 -> (M=lane-16, K=k0+2,k0+3)
// B frag (4x16 f32): lane<16 -> (K=k0,k0+1, N=n0+lane); lane>=16 -> (K=k0+2,k0+3, N=n0+lane-16)
// C (16x16 f32): vgpr r: lanes0-15 = (M=r, N=lane); lanes16-31 = (M=r+8, N=lane-16)
// ---------------------------------------------------------------------------
__global__ __launch_bounds__(256) void gat_gemm_wmma_kernel(
    const float* __restrict__ x, const float* __restrict__ W,
    float* __restrict__ h, int N) {
    __shared__ float sW[64 * 64];
    int t = threadIdx.x;
    for (int i = t * 4; i < 64 * 64; i += 256 * 4)
        *(float4*)(sW + i) = *(const float4*)(W + i);
    __syncthreads();

    int wave = t >> 5;
    int lane = t & 31;
    int tile = blockIdx.x * 8 + wave;              // 16-row tile index (wave-uniform)
    int nTiles = (N + 15) >> 4;
    if (tile >= nTiles) return;                    // wave-uniform exit: EXEC stays full

    int row0   = tile * 16;
    int lhalf  = lane >> 4;                        // 0 or 1
    int l15    = lane & 15;
    int arow   = row0 + l15;
    if (arow >= N) arow = N - 1;                   // clamp reads for a partial last tile
    const float* xrow = x + (size_t)arow * 64;

    v8f c0 = {}, c1 = {}, c2 = {}, c3 = {};
    #pragma unroll
    for (int k0 = 0; k0 < 64; k0 += 4) {
        int ka = k0 + 2 * lhalf;
        v2f a; a.x = xrow[ka]; a.y = xrow[ka + 1];
        const float* wr0 = sW + ka * 64 + l15;
        const float* wr1 = wr0 + 64;
        v2f b0, b1, b2, b3;
        b0.x = wr0[0];  b0.y = wr1[0];
        b1.x = wr0[16]; b1.y = wr1[16];
        b2.x = wr0[32]; b2.y = wr1[32];
        b3.x = wr0[48]; b3.y = wr1[48];
        c0 = __builtin_amdgcn_wmma_f32_16x16x4_f32(false, a, false, b0, (short)0, c0, false, false);
        c1 = __builtin_amdgcn_wmma_f32_16x16x4_f32(false, a, false, b1, (short)0, c1, false, false);
        c2 = __builtin_amdgcn_wmma_f32_16x16x4_f32(false, a, false, b2, (short)0, c2, false, false);
        c3 = __builtin_amdgcn_wmma_f32_16x16x4_f32(false, a, false, b3, (short)0, c3, false, false);
    }

    // store C tiles: element (r) of c_t -> h[row0 + r + 8*lhalf][16*t + l15]
    int rbase = row0 + 8 * lhalf;
    #pragma unroll
    for (int r = 0; r < 8; ++r) {
        int row = rbase + r;
        if (row < N) {
            float* ho = h + (size_t)row * 64 + l15;
            ho[0]  = c0[r];
            ho[16] = c1[r];
            ho[32] = c2[r];
            ho[48] = c3[r];
        }
    }
}

// ---------------------------------------------------------------------------
// alpha_s[n] = dot(h[n], a_src); alpha_d[n] = dot(h[n], a_dst). One wave per node.
// ---------------------------------------------------------------------------
__global__ __launch_bounds__(256) void gat_alpha_kernel(
    const float* __restrict__ h, const float* __restrict__ a_src,
    const float* __restrict__ a_dst, float* __restrict__ as,
    float* __restrict__ ad, int N) {
    long long w = ((long long)blockIdx.x * blockDim.x + threadIdx.x) >> 5;
    int lane = threadIdx.x & 31;
    if (w >= N) return;
    const float* hr = h + w * 64;
    float h0 = hr[lane], h1 = hr[lane + 32];
    float ps = h0 * a_src[lane] + h1 * a_src[lane + 32];
    float pd = h0 * a_dst[lane] + h1 * a_dst[lane + 32];
    #pragma unroll
    for (int off = 16; off > 0; off >>= 1) {
        ps += __shfl_down(ps, off, 32);
        pd += __shfl_down(pd, off, 32);
    }
    if (lane == 0) { as[w] = ps; ad[w] = pd; }
}

// ---------------------------------------------------------------------------
// Pass 1 over edges: e = leaky_relu(as[src]+ad[dst]); segment max via atomicMax.
// k in [0,E) -> real edges; k in [E, E+N) -> self loops.
// ---------------------------------------------------------------------------
__global__ void gat_edge_max_kernel(const long long* __restrict__ ei,
                                    const float* __restrict__ as, const float* __restrict__ ad,
                                    float* __restrict__ e_buf, unsigned* __restrict__ emax,
                                    long long E, long long M) {
    long long k = (long long)blockIdx.x * blockDim.x + threadIdx.x;
    if (k >= M) return;
    long long s, d;
    if (k < E) { s = ei[k]; d = ei[E + k]; } else { s = k - E; d = s; }
    float e = as[s] + ad[d];
    e = (e > 0.0f) ? e : NEG_SLOPE * e;
    e_buf[k] = e;
    atomicMax(&emax[d], enc_f32(e));
}

// ---------------------------------------------------------------------------
// Pass 2: e_exp = exp(e - emax[dst]); denom[dst] += e_exp (atomic).
// ---------------------------------------------------------------------------
__global__ void gat_edge_exp_kernel(const long long* __restrict__ ei,
                                    float* __restrict__ e_buf, const unsigned* __restrict__ emax,
                                    float* __restrict__ denom, long long E, long long M) {
    long long k = (long long)blockIdx.x * blockDim.x + threadIdx.x;
    if (k >= M) return;
    long long d = (k < E) ? ei[E + k] : (k - E);
    float ex = expf(e_buf[k] - dec_f32(emax[d]));
    e_buf[k] = ex;
    atomicAdd(&denom[d], ex);
}

// ---------------------------------------------------------------------------
// Pass 3: out[dst] += (e_exp/denom[dst]) * h[src].  One wave per edge; lane
// handles features lane and lane+32 (coalesced 128B gather + coalesced atomics).
// ---------------------------------------------------------------------------
__global__ __launch_bounds__(256) void gat_scatter_kernel(
    const long long* __restrict__ ei, const float* __restrict__ h,
    const float* __restrict__ e_buf, const float* __restrict__ denom,
    float* __restrict__ out, long long E, long long M) {
    long long w = ((long long)blockIdx.x * blockDim.x + threadIdx.x) >> 5;
    int lane = threadIdx.x & 31;
    if (w >= M) return;
    long long s = 0, d = 0;
    if (lane == 0) {
        if (w < E) { s = ei[w]; d = ei[E + w]; } else { s = w - E; d = s; }
    }
    s = __shfl(s, 0, 32);
    d = __shfl(d, 0, 32);
    float coeff = e_buf[w] / denom[d];
    const float* hs = h + s * 64;
    float* od = out + d * 64;
    atomicAdd(&od[lane],      coeff * hs[lane]);
    atomicAdd(&od[lane + 32], coeff * hs[lane + 32]);
}

// ---------------------------------------------------------------------------
extern "C" void kernel_launch(void* const* d_in, const int* in_sizes, int n_in,
                              void* d_out, int out_size, void* d_ws, size_t ws_size,
                              hipStream_t stream) {
    const float*     x     = (const float*)d_in[0];
    const float*     W     = (const float*)d_in[1];
    const float*     a_src = (const float*)d_in[2];
    const float*     a_dst = (const float*)d_in[3];
    const float*     bias  = (const float*)d_in[4];
    const long long* ei    = (const long long*)d_in[5];

    int       N = in_sizes[0] / 64;
    long long E = (long long)in_sizes[5] / 2;
    long long M = E + (long long)N;
    float* out  = (float*)d_out;

    // workspace carve
    char* ws = (char*)d_ws;
    float*    h     = (float*)ws;    ws += (size_t)N * 64 * sizeof(float);
    float*    as    = (float*)ws;    ws += (size_t)N * sizeof(float);
    float*    ad    = (float*)ws;    ws += (size_t)N * sizeof(float);
    float*    denom = (float*)ws;    ws += (size_t)N * sizeof(float);
    unsigned* emax  = (unsigned*)ws; ws += (size_t)N * sizeof(unsigned);
    float*    e_buf = (float*)ws;

    const int B = 256;
    long long initThreads = (long long)N * 64;
    gat_init_kernel<<<(unsigned)((initThreads + B - 1) / B), B, 0, stream>>>(
        out, bias, denom, emax, N);

    int rowTiles = (N + 15) / 16;
    gat_gemm_wmma_kernel<<<(rowTiles + 7) / 8, B, 0, stream>>>(x, W, h, N);

    long long alphaThreads = (long long)N * 32;
    gat_alpha_kernel<<<(unsigned)((alphaThreads + B - 1) / B), B, 0, stream>>>(
        h, a_src, a_dst, as, ad, N);

    gat_edge_max_kernel<<<(unsigned)((M + B - 1) / B), B, 0, stream>>>(
        ei, as, ad, e_buf, emax, E, M);

    gat_edge_exp_kernel<<<(unsigned)((M + B - 1) / B), B, 0, stream>>>(
        ei, e_buf, emax, denom, E, M);

    long long scatThreads = M * 32;
    gat_scatter_kernel<<<(unsigned)((scatThreads + B - 1) / B), B, 0, stream>>>(
        ei, h, e_buf, denom, out, E, M);
}